// DrugRepurposingHeteroGNN_7670811591143
// MI455X (gfx1250) — compile-verified
//
#include <hip/hip_runtime.h>

#define HID 128

typedef float v2f __attribute__((ext_vector_type(2)));
typedef float v8f __attribute__((ext_vector_type(8)));

// ---------------------------------------------------------------------------
// Degree count: one thread per edge, atomic +1.0 on the destination node.
// ---------------------------------------------------------------------------
__global__ void deg_count_kernel(const long long* __restrict__ dst,
                                 float* __restrict__ deg, int n_edges) {
    int stride = gridDim.x * blockDim.x;
    for (int i = blockIdx.x * blockDim.x + threadIdx.x; i < n_edges; i += stride) {
        unsafeAtomicAdd(&deg[(int)dst[i]], 1.0f);
    }
}

// ---------------------------------------------------------------------------
// Scatter-add: one wave (32 lanes) per edge; lane handles float4 (512B row
// fully coalesced). Atomic f32 adds land in L2 (all buffers are L2-resident).
// ---------------------------------------------------------------------------
__global__ void scatter_add_kernel(const float* __restrict__ x_src,
                                   const long long* __restrict__ src,
                                   const long long* __restrict__ dst,
                                   float* __restrict__ agg, int n_edges) {
    const int lane   = threadIdx.x & 31;
    const int wave   = (blockIdx.x * blockDim.x + threadIdx.x) >> 5;
    const int nwaves = (gridDim.x * blockDim.x) >> 5;
    for (int e = wave; e < n_edges; e += nwaves) {
        const int s = (int)src[e];
        const int d = (int)dst[e];
        const float4 v = *(const float4*)(x_src + (size_t)s * HID + lane * 4);
        float* o = agg + (size_t)d * HID + lane * 4;
        unsafeAtomicAdd(o + 0, v.x);
        unsafeAtomicAdd(o + 1, v.y);
        unsafeAtomicAdd(o + 2, v.z);
        unsafeAtomicAdd(o + 3, v.w);
    }
}

// ---------------------------------------------------------------------------
// mean = agg / max(deg, 1)
// ---------------------------------------------------------------------------
__global__ void mean_div_kernel(float* __restrict__ agg,
                                const float* __restrict__ deg, int n_rows) {
    const int total  = n_rows * HID;
    const int stride = gridDim.x * blockDim.x;
    for (int i = blockIdx.x * blockDim.x + threadIdx.x; i < total; i += stride) {
        float d = deg[i >> 7];                  // i / 128
        agg[i] = agg[i] / fmaxf(d, 1.0f);
    }
}

// ---------------------------------------------------------------------------
// Fused SAGE GEMM: Cout = [Cin +] A1@W1 + A2@W2 + bias, optional ReLU.
// A1/A2: n_rows x 128 (row major), W1/W2: 128 x 128 (row major).
// Block = 256 threads = 8 waves. Block -> 16-row tile; wave w -> cols 16w..16w+15.
// Uses V_WMMA_F32_16X16X4_F32 (fp32 precision kept; GEMM is not the bottleneck).
//
// ISA register layouts (CDNA5 ISA):
//   A 16x4 f32 : lanes 0-15 M=0..15 {VGPR0:K0, VGPR1:K1}; lanes 16-31 {K2,K3}
//   B  4x16 f32: lanes 0-15 N=0..15 {VGPR0:K0, VGPR1:K1}; lanes 16-31 {K2,K3}
//   C/D 16x16  : VGPR r -> M=r (lanes 0-15, N=lane) / M=8+r (lanes 16-31)
// flags: bit0 = accumulate (read Cin), bit1 = ReLU.
// ---------------------------------------------------------------------------
__global__ void sage_gemm_kernel(const float* __restrict__ A1,
                                 const float* __restrict__ W1,
                                 const float* __restrict__ A2,
                                 const float* __restrict__ W2,
                                 const float* __restrict__ bias,
                                 const float* __restrict__ Cin,
                                 float* __restrict__ Cout,
                                 int n_rows, int flags) {
    const int lane = threadIdx.x & 31;
    const int wv   = threadIdx.x >> 5;      // 0..7 -> column tile
    const int hi   = lane >> 4;             // 0 or 1 (K-half of the fragment)
    const int lr   = lane & 15;
    const int row0 = blockIdx.x * 16;
    const int col  = wv * 16 + lr;

    int arow = row0 + lr;                   // clamp loads for ragged last tile
    if (arow >= n_rows) arow = n_rows - 1;

    const float* a1 = A1 + (size_t)arow * HID + 2 * hi;
    const float* a2 = A2 + (size_t)arow * HID + 2 * hi;
    const float* w1 = W1 + (size_t)(2 * hi) * HID + col;
    const float* w2 = W2 + (size_t)(2 * hi) * HID + col;

    v8f acc = {0.f, 0.f, 0.f, 0.f, 0.f, 0.f, 0.f, 0.f};

#pragma unroll
    for (int k = 0; k < HID; k += 4) {
        v2f a = *(const v2f*)(a1 + k);      // A[row][k+2hi], A[row][k+2hi+1]
        v2f b;
        b[0] = w1[(size_t)k * HID];         // W[k+2hi][col]
        b[1] = w1[(size_t)(k + 1) * HID];   // W[k+2hi+1][col]
        acc = __builtin_amdgcn_wmma_f32_16x16x4_f32(false, a, false, b,
                                                    (short)0, acc, false, false);
    }
#pragma unroll
    for (int k = 0; k < HID; k += 4) {
        v2f a = *(const v2f*)(a2 + k);
        v2f b;
        b[0] = w2[(size_t)k * HID];
        b[1] = w2[(size_t)(k + 1) * HID];
        acc = __builtin_amdgcn_wmma_f32_16x16x4_f32(false, a, false, b,
                                                    (short)0, acc, false, false);
    }

    const float bv = bias[col];
#pragma unroll
    for (int r = 0; r < 8; r++) {
        const int m = row0 + r + 8 * hi;
        if (m < n_rows) {
            const size_t o = (size_t)m * HID + col;
            float v = acc[r] + bv;
            if (flags & 1) v += Cin[o];
            if (flags & 2) v = fmaxf(v, 0.0f);
            Cout[o] = v;
        }
    }
}

// ---------------------------------------------------------------------------
// Host orchestration (graph-capture safe: only async ops on `stream`).
// ---------------------------------------------------------------------------
extern "C" void kernel_launch(void* const* d_in, const int* in_sizes, int n_in,
                              void* d_out, int out_size, void* d_ws, size_t ws_size,
                              hipStream_t stream) {
    (void)n_in; (void)out_size; (void)ws_size;

    const float* xd = (const float*)d_in[0];
    const float* xg = (const float*)d_in[1];
    const float* xr = (const float*)d_in[2];
    const long long* e_dg  = (const long long*)d_in[3];
    const long long* e_gd  = (const long long*)d_in[4];
    const long long* e_gdr = (const long long*)d_in[5];
    const long long* e_drg = (const long long*)d_in[6];

    const int ND = in_sizes[0] / HID;
    const int NG = in_sizes[1] / HID;
    const int NR = in_sizes[2] / HID;
    const int Edg  = in_sizes[3] / 2;
    const int Egd  = in_sizes[4] / 2;
    const int Egdr = in_sizes[5] / 2;
    const int Edrg = in_sizes[6] / 2;

    // params flattened in setup_inputs insertion order:
    // params1: dg(Wl,bl,Wr) gd(...) gdr(...) drg(...)  then params2 the same.
    const float* P[24];
    for (int i = 0; i < 24; i++) P[i] = (const float*)d_in[7 + i];
    const float *Wl1_dg  = P[0],  *bl1_dg  = P[1],  *Wr1_dg  = P[2];
    const float *Wl1_gd  = P[3],  *bl1_gd  = P[4],  *Wr1_gd  = P[5];
    const float *Wl1_gdr = P[6],  *bl1_gdr = P[7],  *Wr1_gdr = P[8];
    const float *Wl1_drg = P[9],  *bl1_drg = P[10], *Wr1_drg = P[11];
    const float *Wl2_dg  = P[12], *bl2_dg  = P[13], *Wr2_dg  = P[14];
    const float *Wl2_gd  = P[15], *bl2_gd  = P[16], *Wr2_gd  = P[17];
    const float *Wl2_gdr = P[18], *bl2_gdr = P[19], *Wr2_gdr = P[20];
    const float *Wl2_drg = P[21], *bl2_drg = P[22], *Wr2_drg = P[23];

    // ---- workspace layout (floats) ----
    float* w = (float*)d_ws;
    size_t off = 0;
    float* deg_dg  = w + off; off += NG;                 // dst gene
    float* deg_gd  = w + off; off += ND;                 // dst disease
    float* deg_gdr = w + off; off += NR;                 // dst drug
    float* deg_drg = w + off; off += NG;                 // dst gene
    const size_t mean_off = off;
    float* m_dg  = w + off; off += (size_t)NG * HID;
    float* m_gd  = w + off; off += (size_t)ND * HID;
    float* m_gdr = w + off; off += (size_t)NR * HID;
    float* m_drg = w + off; off += (size_t)NG * HID;
    const size_t zero_all_bytes  = off * sizeof(float);
    const size_t mean_bytes      = (off - mean_off) * sizeof(float);
    float* h_d = w + off; off += (size_t)ND * HID;
    float* h_g = w + off; off += (size_t)NG * HID;
    float* h_r = w + off; off += (size_t)NR * HID;

    // output dict order: disease, gene, drug
    float* out_d = (float*)d_out;
    float* out_g = out_d + (size_t)ND * HID;
    float* out_r = out_g + (size_t)NG * HID;

    const int TB = 256;
    const int gb_d = (ND + 15) / 16;
    const int gb_g = (NG + 15) / 16;
    const int gb_r = (NR + 15) / 16;

    // ---- zero degrees + aggregation buffers ----
    hipMemsetAsync(d_ws, 0, zero_all_bytes, stream);

    // ---- degrees (depend only on dst indices; shared by both layers) ----
    deg_count_kernel<<<1024, TB, 0, stream>>>(e_dg  + Edg,  deg_dg,  Edg);
    deg_count_kernel<<<1024, TB, 0, stream>>>(e_gd  + Egd,  deg_gd,  Egd);
    deg_count_kernel<<<1024, TB, 0, stream>>>(e_gdr + Egdr, deg_gdr, Egdr);
    deg_count_kernel<<<1024, TB, 0, stream>>>(e_drg + Edrg, deg_drg, Edrg);

    // ================= layer 1 =================
    scatter_add_kernel<<<4096, TB, 0, stream>>>(xd, e_dg,  e_dg  + Edg,  m_dg,  Edg);
    scatter_add_kernel<<<4096, TB, 0, stream>>>(xg, e_gd,  e_gd  + Egd,  m_gd,  Egd);
    scatter_add_kernel<<<4096, TB, 0, stream>>>(xg, e_gdr, e_gdr + Egdr, m_gdr, Egdr);
    scatter_add_kernel<<<4096, TB, 0, stream>>>(xr, e_drg, e_drg + Edrg, m_drg, Edrg);

    mean_div_kernel<<<2048, TB, 0, stream>>>(m_dg,  deg_dg,  NG);
    mean_div_kernel<<<2048, TB, 0, stream>>>(m_gd,  deg_gd,  ND);
    mean_div_kernel<<<2048, TB, 0, stream>>>(m_gdr, deg_gdr, NR);
    mean_div_kernel<<<2048, TB, 0, stream>>>(m_drg, deg_drg, NG);

    // gene = relu(conv_dg + conv_drg): two accumulating passes
    sage_gemm_kernel<<<gb_g, TB, 0, stream>>>(m_dg,  Wl1_dg,  xg, Wr1_dg,  bl1_dg,  h_g, h_g, NG, 0);
    sage_gemm_kernel<<<gb_g, TB, 0, stream>>>(m_drg, Wl1_drg, xg, Wr1_drg, bl1_drg, h_g, h_g, NG, 1 | 2);
    sage_gemm_kernel<<<gb_d, TB, 0, stream>>>(m_gd,  Wl1_gd,  xd, Wr1_gd,  bl1_gd,  h_d, h_d, ND, 2);
    sage_gemm_kernel<<<gb_r, TB, 0, stream>>>(m_gdr, Wl1_gdr, xr, Wr1_gdr, bl1_gdr, h_r, h_r, NR, 2);

    // ================= layer 2 =================
    hipMemsetAsync(m_dg, 0, mean_bytes, stream);

    scatter_add_kernel<<<4096, TB, 0, stream>>>(h_d, e_dg,  e_dg  + Edg,  m_dg,  Edg);
    scatter_add_kernel<<<4096, TB, 0, stream>>>(h_g, e_gd,  e_gd  + Egd,  m_gd,  Egd);
    scatter_add_kernel<<<4096, TB, 0, stream>>>(h_g, e_gdr, e_gdr + Egdr, m_gdr, Egdr);
    scatter_add_kernel<<<4096, TB, 0, stream>>>(h_r, e_drg, e_drg + Edrg, m_drg, Edrg);

    mean_div_kernel<<<2048, TB, 0, stream>>>(m_dg,  deg_dg,  NG);
    mean_div_kernel<<<2048, TB, 0, stream>>>(m_gd,  deg_gd,  ND);
    mean_div_kernel<<<2048, TB, 0, stream>>>(m_gdr, deg_gdr, NR);
    mean_div_kernel<<<2048, TB, 0, stream>>>(m_drg, deg_drg, NG);

    // final layer: no ReLU
    sage_gemm_kernel<<<gb_g, TB, 0, stream>>>(m_dg,  Wl2_dg,  h_g, Wr2_dg,  bl2_dg,  out_g, out_g, NG, 0);
    sage_gemm_kernel<<<gb_g, TB, 0, stream>>>(m_drg, Wl2_drg, h_g, Wr2_drg, bl2_drg, out_g, out_g, NG, 1);
    sage_gemm_kernel<<<gb_d, TB, 0, stream>>>(m_gd,  Wl2_gd,  h_d, Wr2_gd,  bl2_gd,  out_d, out_d, ND, 0);
    sage_gemm_kernel<<<gb_r, TB, 0, stream>>>(m_gdr, Wl2_gdr, h_r, Wr2_gdr, bl2_gdr, out_r, out_r, NR, 0);
}